// PCWrapper_35811437314720
// MI455X (gfx1250) — compile-verified
//
#include <hip/hip_runtime.h>
#include <hip/hip_bf16.h>

typedef __attribute__((ext_vector_type(2))) float v2f;
typedef __attribute__((ext_vector_type(8))) float v8f;

#define GEMM_NT 5   // N-tiles per wave (75 = 15 * 5), A fragments reused in regs

// ---------------------------------------------------------------------------
// Kernel 1: h = relu(x_distorted @ w1^T)   (B,5)x(5,128) -> (B,128)
// ---------------------------------------------------------------------------
__global__ void mlp1_kernel(const float* __restrict__ xdist,
                            const float* __restrict__ w1,
                            float* __restrict__ h, int B) {
  int idx = blockIdx.x * 256 + threadIdx.x;
  if (idx >= B * 128) return;
  int b = idx >> 7;
  int j = idx & 127;
  const float* xp = xdist + b * 5;
  const float* wp = w1 + j * 5;
  float s = xp[0] * wp[0];
  s = fmaf(xp[1], wp[1], s);
  s = fmaf(xp[2], wp[2], s);
  s = fmaf(xp[3], wp[3], s);
  s = fmaf(xp[4], wp[4], s);
  h[idx] = fmaxf(s, 0.0f);
}

// ---------------------------------------------------------------------------
// Kernel 2: xd = h @ w2^T + b2   M=B, N=1200, K=128 via V_WMMA_F32_16X16X4_F32
// One wave computes GEMM_NT consecutive 16x16 N-tiles for one M-tile.
// A fragments (32 x float2 = 64 VGPRs) loaded once, reused across all tiles;
// only B streams per tile => ~1.2 loads per WMMA instead of 2.
// Fragment layouts (ISA 7.12.2):
//   A 16x4 f32: lane holds row=lane&15, K-pair base=(lane>>4)*2
//   B 4x16 f32: lane holds col=lane&15, same K addressing vs w2 (B[k][n]=w2[n*128+k])
//   C/D 16x16:  vgpr r -> M = r + (lane>>4)*8, N = lane&15
// ---------------------------------------------------------------------------
__global__ void gemm2_wmma_kernel(const float* __restrict__ h,
                                  const float* __restrict__ w2,
                                  const float* __restrict__ b2,
                                  float* __restrict__ xd,
                                  int mtiles, int ngroups) {
  int wave = threadIdx.x >> 5;
  int lane = threadIdx.x & 31;
  int gwid = blockIdx.x * (blockDim.x >> 5) + wave;
  if (gwid >= mtiles * ngroups) return;   // uniform per wave: EXEC stays all-1s
  int mtile = gwid / ngroups;
  int ngrp  = gwid % ngroups;

  int rc    = lane & 15;    // A-row / B-col / D-col
  int khalf = lane >> 4;    // selects K pair {0,1} vs {2,3}

  // Load all 32 A fragments for this M-tile once.
  const float* aptr = h + (size_t)(mtile * 16 + rc) * 128 + khalf * 2;
  v2f afrag[32];
#pragma unroll
  for (int kk = 0; kk < 32; ++kk) afrag[kk] = *(const v2f*)(aptr + kk * 4);

  int mbase = mtile * 16 + khalf * 8;

  for (int t = 0; t < GEMM_NT; ++t) {
    int ntile = ngrp * GEMM_NT + t;
    const float* bptr = w2 + (size_t)(ntile * 16 + rc) * 128 + khalf * 2;

    v8f acc = {};
#pragma unroll
    for (int kk = 0; kk < 32; ++kk) {
      v2f b = *(const v2f*)(bptr + kk * 4);
      acc = __builtin_amdgcn_wmma_f32_16x16x4_f32(
          /*neg_a=*/false, afrag[kk], /*neg_b=*/false, b,
          /*c_mod=*/(short)0, acc, /*reuse_a=*/false, /*reuse_b=*/false);
    }

    int n = ntile * 16 + rc;
    float bias = b2[n];
#pragma unroll
    for (int r = 0; r < 8; ++r) {
      xd[(size_t)(mbase + r) * 1200 + n] = acc[r] + bias;
    }
  }
}

// ---------------------------------------------------------------------------
// Kernel 3: per batch element: 3x3 SAME conv (6->15 ch, 20x20, zero-padded
// 22x22 LDS tile) + bias, then modified Gram-Schmidt over 5 vectors of 1200,
// all in LDS. One block per image.
// ---------------------------------------------------------------------------
__device__ __forceinline__ float block_dot(const float* a, const float* b,
                                           float* red, float* scal) {
  float part = 0.0f;
  for (int e = threadIdx.x; e < 1200; e += 256) part += a[e] * b[e];
#pragma unroll
  for (int off = 16; off > 0; off >>= 1)
    part += __shfl_down(part, off, 32);
  int lane = threadIdx.x & 31;
  int wid  = threadIdx.x >> 5;
  if (lane == 0) red[wid] = part;
  __syncthreads();
  if (threadIdx.x == 0) {
    float s = 0.0f;
#pragma unroll
    for (int w = 0; w < 8; ++w) s += red[w];
    *scal = s;
  }
  __syncthreads();
  return *scal;
}

__global__ void conv_gs_kernel(const float* __restrict__ xd,
                               const float* __restrict__ xr,
                               const float* __restrict__ cw,
                               const float* __restrict__ cb,
                               float* __restrict__ out) {
  __shared__ float sX[6 * 22 * 22];   // zero-padded input tile
  __shared__ float sW[6000];          // conv output == w_flat (5,1200)
  __shared__ float sCW[810];
  __shared__ float sCB[15];
  __shared__ float sRed[8];
  __shared__ float sScal;

  int b   = blockIdx.x;
  int tid = threadIdx.x;

  for (int i = tid; i < 6 * 484; i += 256) sX[i] = 0.0f;
  for (int i = tid; i < 810; i += 256) sCW[i] = cw[i];
  if (tid < 15) sCB[tid] = cb[tid];
  __syncthreads();

  const float* src0 = xd + (size_t)b * 1200;   // channels 0..2 (from MLP)
  const float* src1 = xr + (size_t)b * 1200;   // channels 3..5 (x_restored)
  for (int i = tid; i < 2400; i += 256) {
    int c = i / 400, p = i % 400, y = p / 20, x = p % 20;
    float v = (c < 3) ? src0[c * 400 + p] : src1[(c - 3) * 400 + p];
    sX[c * 484 + (y + 1) * 22 + (x + 1)] = v;
  }
  __syncthreads();

  // conv: 54 branch-free MACs per output
  for (int o = tid; o < 6000; o += 256) {
    int oc = o / 400, p = o % 400, y = p / 20, x = p % 20;
    float s = sCB[oc];
#pragma unroll
    for (int ic = 0; ic < 6; ++ic) {
      const float* xp = &sX[ic * 484 + y * 22 + x];
      const float* wp = &sCW[(oc * 6 + ic) * 9];
#pragma unroll
      for (int dy = 0; dy < 3; ++dy)
#pragma unroll
        for (int dx = 0; dx < 3; ++dx)
          s = fmaf(xp[dy * 22 + dx], wp[dy * 3 + dx], s);
    }
    sW[o] = s;
  }
  __syncthreads();

  // modified Gram-Schmidt over 5 rows of sW (each length 1200)
  float* ob = out + (size_t)b * 6000;
  for (int i = 0; i < 5; ++i) {
    float* wi = &sW[i * 1200];
    for (int j = 0; j < i; ++j) {
      const float* wj = &sW[j * 1200];          // already normalized
      float d = block_dot(wi, wj, sRed, &sScal);
      for (int e = tid; e < 1200; e += 256) wi[e] -= d * wj[e];
      __syncthreads();
    }
    // emit un-normalized orthogonal vector
    for (int e = tid; e < 1200; e += 256) ob[i * 1200 + e] = wi[e];
    // normalize the LDS copy for use as projector
    float nn  = block_dot(wi, wi, sRed, &sScal);
    float inv = rsqrtf(nn);
    for (int e = tid; e < 1200; e += 256) wi[e] *= inv;
    __syncthreads();
  }
}

// ---------------------------------------------------------------------------
extern "C" void kernel_launch(void* const* d_in, const int* in_sizes, int n_in,
                              void* d_out, int out_size, void* d_ws, size_t ws_size,
                              hipStream_t stream) {
  const float* xr    = (const float*)d_in[0];  // (B,3,20,20)
  const float* xdist = (const float*)d_in[1];  // (B,5)
  const float* w1    = (const float*)d_in[2];  // (128,5)
  const float* w2    = (const float*)d_in[3];  // (1200,128)
  const float* b2    = (const float*)d_in[4];  // (1200,)
  const float* cw    = (const float*)d_in[5];  // (15,6,3,3)
  const float* cb    = (const float*)d_in[6];  // (15,)
  float* out = (float*)d_out;

  int B = in_sizes[1] / 5;                     // 8192

  // workspace: h (B*128 f32) then xd (B*1200 f32)  => ~43.3 MB for B=8192
  float* h  = (float*)d_ws;
  float* xd = h + (size_t)B * 128;

  mlp1_kernel<<<(B * 128 + 255) / 256, 256, 0, stream>>>(xdist, w1, h, B);

  int mtiles  = B / 16;                        // 512
  int ngroups = (1200 / 16) / GEMM_NT;         // 75 / 5 = 15
  int waves   = mtiles * ngroups;              // 7680 waves
  gemm2_wmma_kernel<<<(waves + 7) / 8, 256, 0, stream>>>(h, w2, b2, xd,
                                                         mtiles, ngroups);

  conv_gs_kernel<<<B, 256, 0, stream>>>(xd, xr, cw, cb, out);
}